// SinkhornLabelAllocation_65128884077218
// MI455X (gfx1250) — compile-verified
//
#include <hip/hip_runtime.h>
#include <math.h>

// ---------------------------------------------------------------------------
// Sinkhorn Label Allocation, compressed form for MI455X (gfx1250).
// Active block: 4096 x 101 (L2-resident); untouched interior rows collapse to
// one weighted virtual row; last row is a second virtual row.
// WMMA f32 16x16x4 performs the row-sum-of-exp reductions.
// Scratch requirement: ~3.4 MB of d_ws.
// ---------------------------------------------------------------------------

#define BATCH   4096
#define NCLS    100
#define CP      101          // NCLS + 1
#define NEX     200000
#define REGc    0.5f
#define RHOc    0.5f
#define TOLc    0.01f
#define MAXIT   300
#define NBLK    32           // persistent-kernel grid (guaranteed co-resident)
#define NTHR    256
#define TOUCHW  6256

typedef float v2f __attribute__((ext_vector_type(2)));
typedef float v8f __attribute__((ext_vector_type(8)));

struct Scratch {
  int bar, n_unique, flag, _pad;
  unsigned touch[TOUCHW];                      // touched-row bitmask
  float err, u_rest0, u_last0, c_sum, log_rn, log_nrest, u_rest_out, u_last_out, m0;
  float v_cur[CP], v_raw[CP], log_c[CP], c_arr[CP];
  float col_part[NBLK * CP];                   // per-block column partials
  float u_act[BATCH];
  int   owner[BATCH];
};

#define A_RM_OFS 131072
#define A_CM_OFS (A_RM_OFS + BATCH * CP * 4)   // 1,785,856 ; total ws ~3.44 MB
static_assert(sizeof(Scratch) < A_RM_OFS, "scratch layout overflow");

// ---- software grid barrier (cumulative count, no reset) -------------------
__device__ __forceinline__ void grid_sync(int* bar, int target) {
  __threadfence();
  __syncthreads();
  if (threadIdx.x == 0) {
    __hip_atomic_fetch_add(bar, 1, __ATOMIC_ACQ_REL, __HIP_MEMORY_SCOPE_AGENT);
    while (__hip_atomic_load(bar, __ATOMIC_ACQUIRE, __HIP_MEMORY_SCOPE_AGENT) < target)
      __builtin_amdgcn_s_sleep(1);
  }
  __syncthreads();
}

// ---- K0: zero counters / flags / touch bitmask ----------------------------
__global__ void k_zero(int* p) {
  for (int i = threadIdx.x; i < 8192; i += 256) p[i] = 0;   // first 32 KB
}

// ---- K1: log_softmax, A row-major, q output, initial u for touched rows ---
__global__ void k_prep(const float* __restrict__ logits, const float* __restrict__ vin,
                       float* __restrict__ qout, float* __restrict__ A_rm,
                       Scratch* __restrict__ S) {
  const int wave = threadIdx.x >> 5, lane = threadIdx.x & 31;
  const int b = blockIdx.x * 8 + wave;                 // 512 blocks * 8 waves = 4096
  const float* lrow = logits + (size_t)b * NCLS;

  float mx = -1e30f;
  for (int j = lane; j < NCLS; j += 32) mx = fmaxf(mx, lrow[j]);
  for (int m = 16; m; m >>= 1) mx = fmaxf(mx, __shfl_xor(mx, m));
  float s = 0.f;
  for (int j = lane; j < NCLS; j += 32) s += __expf(lrow[j] - mx);
  for (int m = 16; m; m >>= 1) s += __shfl_xor(s, m);
  const float lse = mx + __logf(s);                     // logsumexp(logits)

  // a_j = 0.5*logp_j (j<100), 0 at j=100 ; z_j = vin_j + a_j
  float zmx = -1e30f;
  for (int j = lane; j < CP; j += 32) {
    float a = (j < NCLS) ? 0.5f * (lrow[j] - lse) : 0.0f;
    A_rm[(size_t)b * CP + j] = a;
    zmx = fmaxf(zmx, vin[j] + a);
  }
  for (int m = 16; m; m >>= 1) zmx = fmaxf(zmx, __shfl_xor(zmx, m));
  float zs = 0.f;
  for (int j = lane; j < CP; j += 32) {
    float a = (j < NCLS) ? 0.5f * (lrow[j] - lse) : 0.0f;
    zs += __expf(vin[j] + a - zmx);
  }
  for (int m = 16; m; m >>= 1) zs += __shfl_xor(zs, m);
  const float zlse = zmx + __logf(zs);
  if (lane == 0) S->u_act[b] = -zlse;                  // initial u at idxs rows
  for (int j = lane; j < NCLS; j += 32)
    qout[(size_t)b * NCLS + j] = __expf(vin[j] + 0.5f * (lrow[j] - lse) - zlse);
}

// ---- K2: owner = last occurrence of each idx; masked column-major copy ----
__global__ void k_owner(const int* __restrict__ idxs, const float* __restrict__ A_rm,
                        float* __restrict__ A_cm, Scratch* __restrict__ S) {
  const int b = blockIdx.x, tid = threadIdx.x;
  const int myidx = idxs[b];
  __shared__ int dup;
  if (tid == 0) dup = 0;
  __syncthreads();
  int d = 0;
  for (int b2 = tid; b2 < BATCH; b2 += NTHR)
    if (b2 > b && idxs[b2] == myidx) d = 1;
  if (d) dup = 1;                                      // benign same-value race
  __syncthreads();
  const int ow = !dup;
  if (tid == 0) {
    if (ow) atomicAdd(&S->n_unique, 1);
    atomicOr(&S->touch[myidx >> 5], 1u << (myidx & 31));
    S->owner[b] = ow;
  }
  for (int j = tid; j < CP; j += NTHR)
    A_cm[(size_t)j * BATCH + b] = ow ? A_rm[(size_t)b * CP + j] : -1e30f;
}

// ---- K3: constants c, log_c, log_rn, seeds for u_rest/u_last --------------
__global__ void k_seed(const float* __restrict__ costm, const float* __restrict__ uin,
                       const float* __restrict__ vin, const float* __restrict__ lub,
                       Scratch* __restrict__ S) {
  const int tid = threadIdx.x;
  __shared__ float red[128];
  red[tid] = (tid < NCLS) ? __expf(lub[tid]) : 0.f;
  __syncthreads();
  for (int s = 64; s; s >>= 1) { if (tid < s) red[tid] += red[tid + s]; __syncthreads(); }
  const float mu = 1.f - red[0];
  __syncthreads();
  float cj = 0.f;
  if (tid < CP) {
    cj = (tid < NCLS) ? (1.f + (float)NEX * __expf(lub[tid]))
                      : (1.f + (float)NEX * (1.f - RHOc + fmaxf(mu, 0.f)));
    S->c_arr[tid] = cj;
    S->log_c[tid] = __logf(cj);
    S->v_cur[tid] = vin[tid];
  }
  red[tid] = cj;
  __syncthreads();
  for (int s = 64; s; s >>= 1) { if (tid < s) red[tid] += red[tid + s]; __syncthreads(); }
  if (tid == 0) {
    S->c_sum = red[0];
    const float rn = 1.f + (float)NCLS + (float)NEX * (1.f - RHOc - fminf(mu, 0.f));
    S->log_rn = __logf(rn);
    S->log_nrest = __logf((float)(NEX - S->n_unique));
    S->m0 = -REGc * costm[0];                          // interior cost value
    int i = 0;                                         // first untouched row (<4097)
    for (int w = 0; w < TOUCHW; ++w) {
      unsigned t = S->touch[w];
      if (~t) { i = w * 32 + __ffs(~t) - 1; break; }
    }
    S->u_rest0 = uin[i];  S->u_rest_out = uin[i];
    S->u_last0 = uin[NEX]; S->u_last_out = uin[NEX];
    S->flag = 0; S->err = 1e30f;
  }
}

// ---- persistent Sinkhorn loop kernel --------------------------------------
__global__ void k_sink(Scratch* __restrict__ S, const float* __restrict__ A_rm,
                       const float* __restrict__ A_cm) {
  const int tid = threadIdx.x, bid = blockIdx.x;
  const int lane = tid & 31, wave = tid >> 5;
  const int half = lane >> 4, m = lane & 15;
  __shared__ float red[NTHR];
  __shared__ float vsh[CP];
  __shared__ float colacc[CP];
  __shared__ float ush[128];

  float u_rest = S->u_rest0, u_last = S->u_last0;
  const float m0 = S->m0, log_rn = S->log_rn, log_nrest = S->log_nrest;
  const float c_sum_inv = 1.f / S->c_sum;
  float* u_act = S->u_act;
  int gen = 0;

  for (int it = 0; it < MAXIT; ++it) {
    // ---- Phase A: v_raw[j] = log_c[j] - logsumexp_rows(mat_col_j + u) ----
    for (int k = 0; k < 4; ++k) {
      const int j = bid + 32 * k;                      // uniform per block
      if (j < CP) {
        const float* col = A_cm + (size_t)j * BATCH;
        float s = 0.f;
        for (int b = tid; b < BATCH; b += NTHR) {
          __builtin_prefetch(col + b + 2048, 0, 1);    // global_prefetch_b8
          s += __expf(col[b] + u_act[b]);              // masked rows -> exp(-1e30)=0
        }
        red[tid] = s; __syncthreads();
        for (int st = NTHR / 2; st; st >>= 1) { if (tid < st) red[tid] += red[tid + st]; __syncthreads(); }
        if (tid == 0) {
          const float rest = __expf(log_nrest + ((j < NCLS) ? m0 : 0.f) + u_rest);
          S->v_raw[j] = S->log_c[j] - __logf(red[0] + rest + __expf(u_last));
        }
        __syncthreads();
      }
    }
    grid_sync(&S->bar, (++gen) * NBLK);

    // ---- Phase A2: center v, compute u_rest/u_last (replicated per block) -
    if (tid < CP) vsh[tid] = S->v_raw[tid];
    __syncthreads();
    red[tid] = (tid < NCLS) ? vsh[tid] : 0.f; __syncthreads();
    for (int st = NTHR / 2; st; st >>= 1) { if (tid < st) red[tid] += red[tid + st]; __syncthreads(); }
    const float vmean = red[0] * (1.f / (float)NCLS);
    __syncthreads();
    if (tid < CP) vsh[tid] -= vmean;
    __syncthreads();
    if (bid == 0 && tid < CP) S->v_cur[tid] = vsh[tid];
    float er = 0.f, el = 0.f;
    if (tid < CP) {
      const float vv = vsh[tid];
      er = __expf(((tid < NCLS) ? m0 : 0.f) + vv);
      el = __expf(vv);
    }
    red[tid] = er; __syncthreads();
    for (int st = NTHR / 2; st; st >>= 1) { if (tid < st) red[tid] += red[tid + st]; __syncthreads(); }
    const float u_rest_new = -__logf(red[0]);
    __syncthreads();
    red[tid] = el; __syncthreads();
    for (int st = NTHR / 2; st; st >>= 1) { if (tid < st) red[tid] += red[tid + st]; __syncthreads(); }
    const float u_last_new = -__logf(red[0]) + log_rn;
    __syncthreads();

    // ---- Phase B: row logsumexp via WMMA f32 16x16x4 + column partials ----
    if (tid < CP) colacc[tid] = 0.f;
    __syncthreads();
    {
      const int r0 = bid * 128 + wave * 16;            // 32 blk * 8 waves * 16 = 4096 rows
      const int row = r0 + m;
      const float* arow = A_rm + (size_t)row * CP;
      v8f acc = {0.f, 0.f, 0.f, 0.f, 0.f, 0.f, 0.f, 0.f};
      const v2f ones = {1.f, 1.f};
      for (int jc = 0; jc < 104; jc += 4) {            // K-chunks of 4 columns
        const int j0 = jc + 2 * half;
        const float x0 = (j0     < CP) ? __expf(arow[j0]     + vsh[j0])     : 0.f;
        const float x1 = (j0 + 1 < CP) ? __expf(arow[j0 + 1] + vsh[j0 + 1]) : 0.f;
        const v2f af = {x0, x1};
        // D[m][n] = sum_k A[m][k]*1 + C : per-row sum-of-exp, f32 exact
        acc = __builtin_amdgcn_wmma_f32_16x16x4_f32(false, af, false, ones,
                                                    (short)0, acc, false, false);
      }
      // D layout: VGPR i -> rows i (lanes 0-15) / 8+i (lanes 16-31)
      if (lane == 0 || lane == 16) {
        const int rb = (lane >> 4) << 3;
#pragma unroll
        for (int i = 0; i < 8; ++i) {
          const float uu = -__logf(acc[i]);
          u_act[r0 + rb + i] = uu;
          ush[wave * 16 + rb + i] = uu;
        }
      }
      __syncthreads();
      const float ow = (float)S->owner[row];
      const float w0 = __expf(ush[wave * 16 + m]) * ow;   // exp(u_row) = 1/rowsum
      for (int jc = 0; jc < 104; jc += 4) {
        const int j0 = jc + 2 * half;
        if (j0     < CP) atomicAdd(&colacc[j0],     __expf(arow[j0]     + vsh[j0])     * w0);
        if (j0 + 1 < CP) atomicAdd(&colacc[j0 + 1], __expf(arow[j0 + 1] + vsh[j0 + 1]) * w0);
      }
    }
    __syncthreads();
    if (tid < CP) S->col_part[bid * CP + tid] = colacc[tid];
    grid_sync(&S->bar, (++gen) * NBLK);

    // ---- Phase C (block 0): error and convergence flag --------------------
    if (bid == 0) {
      float diff = 0.f;
      if (tid < CP) {
        float Q = 0.f;
        for (int bb = 0; bb < NBLK; ++bb) Q += S->col_part[bb * CP + tid]; // fixed order
        const float vv = vsh[tid];
        Q += __expf(log_nrest + ((tid < NCLS) ? m0 : 0.f) + u_rest_new + vv);
        Q += __expf(u_last_new + vv);
        diff = fabsf(Q - S->c_arr[tid]);
      }
      red[tid] = diff; __syncthreads();
      for (int st = NTHR / 2; st; st >>= 1) { if (tid < st) red[tid] += red[tid + st]; __syncthreads(); }
      if (tid == 0) {
        const float err = red[0] * c_sum_inv;
        S->err = err;
        S->flag = (err >= TOLc) ? 0 : 1;               // NaN => stop (matches while-cond)
        S->u_rest_out = u_rest_new;
        S->u_last_out = u_last_new;
      }
    }
    grid_sync(&S->bar, (++gen) * NBLK);
    u_rest = u_rest_new;
    u_last = u_last_new;
    if (__hip_atomic_load(&S->flag, __ATOMIC_ACQUIRE, __HIP_MEMORY_SCOPE_AGENT)) break;
  }
}

// ---- K5: stream-fill log_Q with rest/last pattern (81 MB, HBM-bound) ------
__global__ void k_fill(const Scratch* __restrict__ S, float* __restrict__ logq) {
  const int i = blockIdx.x, j = threadIdx.x;
  if (j >= CP) return;
  const float v = S->v_cur[j];
  const float val = (i == NEX) ? (v + S->u_last_out)
                               : (((j < NCLS) ? S->m0 : 0.f) + v + S->u_rest_out);
  logq[(size_t)i * CP + j] = val;
}

// ---- K6: overwrite touched rows (owners only) + err scalar ----------------
__global__ void k_scatter(const Scratch* __restrict__ S, const float* __restrict__ A_rm,
                          const int* __restrict__ idxs, float* __restrict__ logq,
                          float* __restrict__ erro) {
  const int b = blockIdx.x, j = threadIdx.x;
  if (b == 0 && j == 0) *erro = S->err;
  if (!S->owner[b] || j >= CP) return;
  const size_t row = (size_t)idxs[b];
  logq[row * CP + j] = A_rm[(size_t)b * CP + j] + S->v_cur[j] + S->u_act[b];
}

// ---------------------------------------------------------------------------
extern "C" void kernel_launch(void* const* d_in, const int* in_sizes, int n_in,
                              void* d_out, int out_size, void* d_ws, size_t ws_size,
                              hipStream_t stream) {
  const float* logits = (const float*)d_in[0];
  const float* costm  = (const float*)d_in[1];
  const float* u_in   = (const float*)d_in[2];
  const float* v_in   = (const float*)d_in[3];
  const float* lub    = (const float*)d_in[4];
  const int*   idxs   = (const int*)d_in[5];
  (void)in_sizes; (void)n_in; (void)out_size; (void)ws_size;

  float* out   = (float*)d_out;
  float* q_out = out;                                        // [4096,100]
  float* logq  = out + (size_t)BATCH * NCLS;                 // [200001,101]
  float* erro  = out + (size_t)BATCH * NCLS + (size_t)(NEX + 1) * CP;

  Scratch* S  = (Scratch*)d_ws;
  float* A_rm = (float*)((char*)d_ws + A_RM_OFS);
  float* A_cm = (float*)((char*)d_ws + A_CM_OFS);

  k_zero   <<<1,        256, 0, stream>>>((int*)d_ws);
  k_prep   <<<512,      256, 0, stream>>>(logits, v_in, q_out, A_rm, S);
  k_owner  <<<BATCH,   NTHR, 0, stream>>>(idxs, A_rm, A_cm, S);
  k_seed   <<<1,        128, 0, stream>>>(costm, u_in, v_in, lub, S);
  k_sink   <<<NBLK,    NTHR, 0, stream>>>(S, A_rm, A_cm);
  k_fill   <<<NEX + 1,  128, 0, stream>>>(S, logq);
  k_scatter<<<BATCH,    128, 0, stream>>>(S, A_rm, idxs, logq, erro);
}